// InterpolationNetwork_9620726743117
// MI455X (gfx1250) — compile-verified
//
#include <hip/hip_runtime.h>
#include <math.h>

typedef __attribute__((ext_vector_type(2))) float v2f;
typedef __attribute__((ext_vector_type(8))) float v8f;

#define BB 64
#define FF 12
#define TT 288
#define RR 192
#define KAPPAC 10.0f

__device__ __forceinline__ float sigmoidf_(float v) {
    return 1.0f / (1.0f + expf(-v));
}

// -------- Kernel 1: single-channel double softmax over T, evaluated at R points
// One block per (b,f), 192 threads (one per r). Output h = sigmoid(stack(y,w,y_trans))
// written directly into d_out (same shape as final output).
__global__ __launch_bounds__(RR) void single_channel_kernel(
    const float* __restrict__ x, const float* __restrict__ kern,
    float* __restrict__ out)
{
    __shared__ float xt[TT];   // x_t
    __shared__ float dd[TT];   // d
    __shared__ float lm[TT];   // log(m)
    __shared__ float al[TT];   // softplus(kernel)
    const int bf  = blockIdx.x;          // b*FF + f
    const int b   = bf / FF;
    const int f   = bf % FF;
    const int tid = threadIdx.x;

    const float* xb = x + ((size_t)(b * 3 + 0) * FF + f) * TT;
    const float* db = x + ((size_t)(b * 3 + 1) * FF + f) * TT;
    const float* mb = x + ((size_t)(b * 3 + 2) * FF + f) * TT;
    for (int i = tid; i < TT; i += RR) {
        xt[i] = xb[i];
        dd[i] = db[i];
        lm[i] = logf(mb[i]);
        al[i] = log1pf(expf(kern[i]));   // softplus
    }
    __syncthreads();

    const int   r   = tid;
    const float ref = (float)RR * (float)r / (float)(RR - 1);  // linspace(0,R,R)

    // pass 1: maxima for the two logsumexps
    float mx = -INFINITY, mxk = -INFINITY;
    for (int t = 0; t < TT; ++t) {
        float df = dd[t] - ref;
        float an = al[t] * df * df;
        float s  = lm[t] - an;
        float sk = lm[t] - KAPPAC * an;
        mx  = fmaxf(mx, s);
        mxk = fmaxf(mxk, sk);
    }
    // pass 2: sums
    float sw = 0.f, swx = 0.f, swk = 0.f, swxk = 0.f;
    for (int t = 0; t < TT; ++t) {
        float df = dd[t] - ref;
        float an = al[t] * df * df;
        float e  = expf((lm[t] - an) - mx);
        float ek = expf((lm[t] - KAPPAC * an) - mxk);
        sw   += e;   swx  += e  * xt[t];
        swk  += ek;  swxk += ek * xt[t];
    }
    float y   = swx  / sw;
    float w   = mx + logf(sw);
    float ytr = swxk / swk;

    size_t base = ((size_t)(b * 3) * FF + f) * RR + r;
    out[base]                        = sigmoidf_(y);
    out[base + (size_t)FF * RR]      = sigmoidf_(w);
    out[base + (size_t)2 * FF * RR]  = sigmoidf_(ytr);
}

// -------- Kernel 2: cross-channel. One block per batch, 192 threads = 6 waves.
// rep = (softmax_F(w) * (y^T - mean)) @ cross_mix + mean, via V_WMMA_F32_16X16X4_F32.
__global__ __launch_bounds__(RR) void cross_channel_kernel(
    const float* __restrict__ cm, float* __restrict__ out)
{
    __shared__ float yl[FF][RR];       // y values [f][r] for the mean reduction
    __shared__ float amat[RR][17];     // A matrix rows, padded cols (12..15 zero)
    __shared__ float cml[FF][16];      // cross_mix, zero-padded to 16 columns
    __shared__ float meanl[16];
    const int b   = blockIdx.x;
    const int tid = threadIdx.x;
    const int r   = tid;

    // stage cross_mix into padded LDS: 12*16 == 192 == blockDim, one elem/thread
    {
        const int kk = tid >> 4;      // 0..11
        const int nn = tid & 15;      // 0..15
        cml[kk][nn] = (nn < FF) ? cm[kk * FF + nn] : 0.f;
    }

    float yv[FF], wv[FF];
    size_t base0 = ((size_t)(b * 3 + 0) * FF) * RR + r;
    size_t base1 = ((size_t)(b * 3 + 1) * FF) * RR + r;
    for (int f = 0; f < FF; ++f) {
        yv[f] = out[base0 + (size_t)f * RR];
        wv[f] = out[base1 + (size_t)f * RR];
    }
    // intensity = sigmoid(exp(w)); only this thread ever touches these addresses
    for (int f = 0; f < FF; ++f)
        out[base1 + (size_t)f * RR] = sigmoidf_(expf(wv[f]));

    // softmax over F (per r)
    float wmax = wv[0];
    for (int f = 1; f < FF; ++f) wmax = fmaxf(wmax, wv[f]);
    float wn[FF];
    float se = 0.f;
    for (int f = 0; f < FF; ++f) { wn[f] = expf(wv[f] - wmax); se += wn[f]; }
    float inv = 1.0f / se;

    for (int f = 0; f < FF; ++f) yl[f][r] = yv[f];
    __syncthreads();

    // mean over R per f
    if (tid < 16) {
        float s = 0.f;
        if (tid < FF)
            for (int rr = 0; rr < RR; ++rr) s += yl[tid][rr];
        meanl[tid] = s * (1.0f / (float)RR);
    }
    __syncthreads();

    // A[r][f] = wn * (y - mean); pad to 16 columns with zeros
    for (int f = 0; f < FF; ++f)
        amat[r][f] = wn[f] * inv * (yv[f] - meanl[f]);
    for (int f = FF; f < 16; ++f)
        amat[r][f] = 0.f;
    __syncthreads();

    // ---- WMMA phase: 12 row-tiles of 16, 6 waves -> 2 tiles each.
    const int wave = tid >> 5;
    const int lane = tid & 31;
    const int half = lane >> 4;   // 0|1 : lanes 16-31 hold K+2 (A,B) / M+8 (D)
    const int l16  = lane & 15;

    // B fragments are tile-invariant: load once from padded LDS, keep in VGPRs.
    // B 4x16 f32 layout: lane L, vgpr v -> N = L%16, K = (L/16)*2 + v
    v2f bfrag[3];
#pragma unroll
    for (int c = 0; c < 3; ++c) {
        const int k0 = c * 4 + half * 2;
        bfrag[c].x = cml[k0][l16];
        bfrag[c].y = cml[k0 + 1][l16];
    }
    const float meanc = meanl[l16];   // output-column mean (cols >= FF unused)

    for (int tile = wave; tile < RR / 16; tile += 6) {
        v8f acc = {};
#pragma unroll
        for (int c = 0; c < 3; ++c) {        // K = 12 in chunks of 4
            const int k0 = c * 4 + half * 2;
            // A 16x4 f32 layout: lane L, vgpr v -> M = L%16, K = (L/16)*2 + v
            v2f a;
            a.x = amat[tile * 16 + l16][k0];
            a.y = amat[tile * 16 + l16][k0 + 1];
            acc = __builtin_amdgcn_wmma_f32_16x16x4_f32(
                false, a, false, bfrag[c], (short)0, acc, false, false);
        }
        // D 16x16 f32 layout: lane L, vgpr v -> N = L%16, M = v + (L/16)*8
        const int fcol = l16;
        if (fcol < FF) {
#pragma unroll
            for (int v = 0; v < 8; ++v) {
                int   rout = tile * 16 + v + half * 8;
                float rep  = acc[v] + meanc;
                size_t i0 = ((size_t)(b * 3 + 0) * FF + fcol) * RR + rout;
                size_t i2 = ((size_t)(b * 3 + 2) * FF + fcol) * RR + rout;
                float h2 = out[i2];
                out[i0] = sigmoidf_(rep);
                out[i2] = sigmoidf_(h2 - rep);
            }
        }
    }
}

extern "C" void kernel_launch(void* const* d_in, const int* in_sizes, int n_in,
                              void* d_out, int out_size, void* d_ws, size_t ws_size,
                              hipStream_t stream) {
    const float* x    = (const float*)d_in[0];   // (B,3,F,T)
    const float* kern = (const float*)d_in[1];   // (T,)
    const float* cm   = (const float*)d_in[2];   // (F,F)
    float* out = (float*)d_out;                  // (B,3,F,R)

    single_channel_kernel<<<dim3(BB * FF), dim3(RR), 0, stream>>>(x, kern, out);
    cross_channel_kernel<<<dim3(BB), dim3(RR), 0, stream>>>(cm, out);
}